// FocalLoss_80719615361797
// MI455X (gfx1250) — compile-verified
//
#include <hip/hip_runtime.h>
#include <hip/hip_bf16.h>
#include <stdint.h>

// Problem constants (from reference)
#define B_N   4
#define A_N   66000
#define N_N   30
#define T_N   5
#define CT    278            // 1 objectness col + 277 class cols
#define ROWS  31             // N+1 table rows
#define TAB_ELEMS (ROWS*CT)  // 8618 floats = 34,472 B per batch
#define PAIRS (CT/2)         // 139 float2 per anchor row
#define WS_TAB_OFF 8         // float offset of table in workspace
#define BLK   256            // 8 wave32 per block
#define GRIDX 512            // blocks per batch for cls_kernel

typedef float        f32x2 __attribute__((ext_vector_type(2)));
typedef float        f32x4 __attribute__((ext_vector_type(4)));
typedef unsigned int u32x4 __attribute__((ext_vector_type(4)));
typedef int          i32x4 __attribute__((ext_vector_type(4)));
typedef int          i32x8 __attribute__((ext_vector_type(8)));

__device__ __forceinline__ float fast_rcp(float x) {
#if defined(__HIP_DEVICE_COMPILE__)
  return __builtin_amdgcn_rcpf(x);
#else
  return 1.0f / x;
#endif
}

// Focal element for binary label y in {0,1}:
//   loss = w(y) * softplus(z) * sigmoid(z)^2,  z = (1-2y)*x,  w = y?ALPHA:1-ALPHA
// Exactly 1 exp + 1 log + 1 rcp per element.
__device__ __forceinline__ float focal01(float x, float y) {
  float z   = (1.0f - 2.0f * y) * x;
  float w   = 0.75f - 0.5f * y;          // ALPHA = 0.25
  float e   = __expf(-fabsf(z));         // e = exp(-|z|) in (0,1]
  float inv = fast_rcp(1.0f + e);
  float sig = (z >= 0.0f) ? inv : e * inv;          // sigmoid(z), stable
  float sp  = fmaxf(z, 0.0f) + __logf(1.0f + e);    // softplus(z), stable
  return w * sp * (sig * sig);
}

// wave32 tree reduction (ds_bpermute path, no LDS memory traffic)
__device__ __forceinline__ float wave_reduce(float v) {
#pragma unroll
  for (int off = 16; off > 0; off >>= 1) v += __shfl_down(v, off, 32);
  return v;
}

// Block-wide reduce-add of v into *gdst:
// shfl tree per wave -> 1 LDS atomic per wave -> 1 global atomic per block.
__device__ __forceinline__ void block_add_to(float* gdst, float v) {
  __shared__ float s;
  if (threadIdx.x == 0) s = 0.0f;
  __syncthreads();
  float wv = wave_reduce(v);
  if ((threadIdx.x & 31) == 0) atomicAdd(&s, wv);
  __syncthreads();
  if (threadIdx.x == 0) atomicAdd(gdst, s);
  __syncthreads();
}

// ---------------- kernel 0: zero accumulators ----------------
__global__ void init_kernel(float* accum) {
  if (threadIdx.x < 4) accum[threadIdx.x] = 0.0f;
}

// ---------------- kernel 1: build per-batch label table ----------------
// tab[b][d][0] = (d>0);  tab[b][d][1+off[g]+c] = one-hot class hits of gt row d-1
__global__ void __launch_bounds__(BLK) table_kernel(const int* __restrict__ gtl,
                                                    float* __restrict__ tab) {
  const int b = blockIdx.x;
  float* t = tab + (size_t)b * TAB_ELEMS;
  for (int k = threadIdx.x; k < TAB_ELEMS; k += blockDim.x) {
    int row = k / CT, col = k - row * CT;
    t[k] = (col == 0 && row > 0) ? 1.0f : 0.0f;
  }
  __syncthreads();
  if (threadIdx.x < N_N * 5) {
    const int off[5] = {0, 10, 33, 102, 265};
    const int ncs[5] = {10, 23, 69, 163, 12};
    int n = threadIdx.x / 5, g = threadIdx.x % 5;
    const int* src = gtl + (((size_t)b * N_N + n) * 5 + g) * T_N;
    for (int tt = 0; tt < T_N; ++tt) {            // cumprod(valid): stop at first <0
      int v = src[tt];
      if (v < 0) break;
      int c = v < ncs[g] - 1 ? v : ncs[g] - 1;
      t[(n + 1) * CT + 1 + off[g] + c] = 1.0f;    // min(sum,1) == idempotent 1.0
    }
  }
}

// ---------------- kernel 2: num_pos + smooth-L1 regression ----------------
__global__ void __launch_bounds__(BLK) posreg_kernel(const int* __restrict__ lbin,
                                                     const float* __restrict__ pred,
                                                     const float* __restrict__ gt,
                                                     float* accum) {
  const unsigned M = B_N * A_N;
  const unsigned stride = blockDim.x * gridDim.x;
  float cnt = 0.0f, reg = 0.0f;
  for (unsigned i = blockIdx.x * blockDim.x + threadIdx.x; i < M; i += stride) {
    int lb = lbin[i];
    if (lb > 0) {
      cnt += 1.0f;
      f32x4 p = *(const f32x4*)(pred + 4u * (size_t)i);
      f32x4 g = *(const f32x4*)(gt   + 4u * (size_t)i);
#pragma unroll
      for (int c = 0; c < 4; ++c) {
        float d = fabsf(p[c] - g[c]);
        // BETA = 1/9: 0.5*d*d/BETA = 4.5*d*d ; d - 0.5*BETA = d - 1/18
        reg += (d < (1.0f / 9.0f)) ? 4.5f * d * d : d - (0.5f / 9.0f);
      }
    }
  }
  block_add_to(accum + 0, cnt);
  block_add_to(accum + 1, reg);
}

// ---------------- kernel 3: class focal loss (dominant, 293 MB stream) ----------------
__global__ void __launch_bounds__(BLK) cls_kernel(const float* __restrict__ conf,
                                                  const int* __restrict__ lbin,
                                                  const float* __restrict__ tab,
                                                  float* accum) {
  const int b = blockIdx.y;
  __shared__ float tbl[TAB_ELEMS];                 // 34,472 B
  const float* gtab = tab + (size_t)b * TAB_ELEMS;

#if defined(__HIP_DEVICE_COMPILE__) && __has_builtin(__builtin_amdgcn_tensor_load_to_lds)
  // CDNA5 Tensor Data Mover: one DMA descriptor pulls the whole per-batch
  // table into LDS. D# per CDNA5 ISA (1-D tile of 8618 dwords).
  if (threadIdx.x == 0) {
    unsigned lds_off = (unsigned)(uintptr_t)&tbl[0];           // addr[31:0] = LDS offset
    unsigned long long ga = (unsigned long long)(uintptr_t)gtab;
    u32x4 g0 = { 1u,                                           // count=1 valid descriptor
                 lds_off,                                      // lds_addr
                 (unsigned)(ga & 0xFFFFFFFFull),               // global_addr[31:0]
                 ((unsigned)((ga >> 32) & 0x1FFFFFFull)) | 0x80000000u }; // [56:32] | type=2
    i32x8 g1 = { 0x00020000,                                   // data_size=2 (4B), mask=0
                 (int)((TAB_ELEMS & 0xFFFF) << 16),            // tensor_dim0 lo16
                 (int)(((TAB_ELEMS >> 16) & 0xFFFF) | (1 << 16)), // dim0 hi16 | tensor_dim1=1
                 (int)((TAB_ELEMS & 0xFFFF) << 16),            // tile_dim0 = 8618
                 0,                                            // tile_dim1=0 (1-D), tile_dim2=0
                 (int)TAB_ELEMS,                               // tensor_dim0_stride lo32
                 0, 0 };
    i32x4 z4 = {0, 0, 0, 0};                                   // groups 2/3 unused (<=2D)
    i32x8 z8 = {0, 0, 0, 0, 0, 0, 0, 0};                       // extra group unused
    __builtin_amdgcn_tensor_load_to_lds(g0, g1, z4, z4, z8, 0);
    __builtin_amdgcn_s_wait_tensorcnt(0);
  }
#else
  for (int k = threadIdx.x; k < TAB_ELEMS; k += blockDim.x) tbl[k] = gtab[k];
#endif
  __syncthreads();

  const f32x2* __restrict__ cp = (const f32x2*)(conf + (size_t)b * A_N * CT);
  const int*   __restrict__ lp = lbin + (size_t)b * A_N;
  const unsigned M = A_N * PAIRS;                  // 9,174,000 float2 elements / batch
  const unsigned stride = blockDim.x * gridDim.x;
  float acc = 0.0f;
  for (unsigned i = blockIdx.x * blockDim.x + threadIdx.x; i < M; i += stride) {
    unsigned a = i / PAIRS;                        // anchor
    unsigned j = i - a * PAIRS;                    // column pair
    int lb = lp[a];                                // broadcast within wave (L2-hot)
    __builtin_prefetch(&cp[i + stride], 0, 0);     // global_prefetch_b8, next stride
    if (lb > -1) {                                 // mask
      f32x2 x = __builtin_nontemporal_load(&cp[i]);            // single-use stream: NT
      int d = lb > 0 ? lb : 0;                                 // dumy row
      f32x2 y = *(const f32x2*)&tbl[(unsigned)d * CT + 2u * j]; // ds_load_b64
      acc += focal01(x.x, y.x) + focal01(x.y, y.y);
    }
  }
  block_add_to(accum + 2, acc);
}

// ---------------- kernel 4: finalize ----------------
__global__ void final_kernel(const float* accum, float* out) {
  float np = fmaxf(1.0f, accum[0]);
  out[0] = accum[1] / (np * 4.0f);   // regression_loss
  out[1] = accum[2] / np;            // cls_loss
}

extern "C" void kernel_launch(void* const* d_in, const int* in_sizes, int n_in,
                              void* d_out, int out_size, void* d_ws, size_t ws_size,
                              hipStream_t stream) {
  const float* conf = (const float*)d_in[0];   // (B,A,278)
  const float* pred = (const float*)d_in[1];   // (B,A,4)
  const float* gt   = (const float*)d_in[2];   // (B,A,4)
  const int*   gtl  = (const int*)d_in[3];     // (B,N,5,T)
  const int*   lbin = (const int*)d_in[4];     // (B,A)
  float* accum = (float*)d_ws;                 // [0]=pos_cnt [1]=reg_sum [2]=cls_sum
  float* tab   = accum + WS_TAB_OFF;           // B * 8618 floats (~138 KB total ws)
  float* out   = (float*)d_out;

  init_kernel<<<1, 32, 0, stream>>>(accum);
  table_kernel<<<B_N, BLK, 0, stream>>>(gtl, tab);
  posreg_kernel<<<256, BLK, 0, stream>>>(lbin, pred, gt, accum);
  cls_kernel<<<dim3(GRIDX, B_N), BLK, 0, stream>>>(conf, lbin, tab, accum);
  final_kernel<<<1, 1, 0, stream>>>(accum, out);
}